// DiffeqSolver_60327110639970
// MI455X (gfx1250) — compile-verified
//
#include <hip/hip_runtime.h>
#include <cmath>

typedef __attribute__((ext_vector_type(2))) float v2f;
typedef __attribute__((ext_vector_type(8))) float v8f;

#define NPTS   100000
#define DIM    64
#define HID    256
#define TSTEPS 20
#define TILES  (NPTS / 16)      // 6250 row-tiles of 16
#define WPB    4                // waves per block
#define THREADS (WPB * 32)

// padded LDS strides (floats): chosen so every b64 fragment load is conflict-free
#define W1T_S 68                // W1T[n][k], n<256, k<64
#define W2T_S 260               // W2T[n][k], n<64,  k<256
#define Y_S   68                // per-wave y staging, 16 x 64
#define HS_S  20                // per-wave h-tile slot, 16 x 16

struct Lds {
  float w1t[HID * W1T_S];       // 17408 f
  float w2t[DIM * W2T_S];       // 16640 f
  float b1[HID];                //   256 f
  float b2[DIM];                //    64 f
  float ybuf[WPB][16 * Y_S];    //  4352 f
  float hbuf[WPB][16 * HS_S];   //  1280 f
};                              // total = 40000 f = 160000 B

__device__ __forceinline__ float fast_tanh(float x) {
#if __has_builtin(__builtin_amdgcn_tanhf)
  return __builtin_amdgcn_tanhf(x);              // v_tanh_f32
#elif __has_builtin(__builtin_amdgcn_tanh_f32)
  return __builtin_amdgcn_tanh_f32(x);
#else
  float e = __expf(-2.0f * fabsf(x));            // v_exp_f32 path, stable for all x
  float t = 1.0f - 2.0f * e / (1.0f + e);
  return copysignf(t, x);
#endif
}

__global__ __launch_bounds__(THREADS)
void node_rk4(const float* __restrict__ y0,
              const float* __restrict__ ts,
              const float* __restrict__ W1,
              const float* __restrict__ b1g,
              const float* __restrict__ W2,
              const float* __restrict__ b2g,
              float* __restrict__ out)
{
  __shared__ Lds lds;
  const int tid = threadIdx.x;

  // Cooperative staging: transpose weights into LDS once per block.
  for (int i = tid; i < DIM * HID; i += THREADS) {   // W1 row-major [64][256]
    int k = i / HID, n = i % HID;
    lds.w1t[n * W1T_S + k] = W1[i];
  }
  for (int i = tid; i < HID * DIM; i += THREADS) {   // W2 row-major [256][64]
    int k = i / DIM, n = i % DIM;
    lds.w2t[n * W2T_S + k] = W2[i];
  }
  for (int i = tid; i < HID; i += THREADS) lds.b1[i] = b1g[i];
  for (int i = tid; i < DIM; i += THREADS) lds.b2[i] = b2g[i];
  __syncthreads();

  const int wave = tid >> 5;
  const int lane = tid & 31;
  const int tile = blockIdx.x * WPB + wave;
  if (tile >= TILES) return;                 // wave-uniform: EXEC stays all-1 for WMMA

  const int l16 = lane & 15;
  const int hi  = lane >> 4;                 // 0/1 lane half
  const int row0 = tile * 16;

  float* __restrict__ Y  = lds.ybuf[wave];
  float* __restrict__ HS = lds.hbuf[wave];

  // Load y0 into WMMA C-layout registers; also emit out[t=0] = y0.
  v8f y[4];
  #pragma unroll
  for (int c = 0; c < 4; ++c)
    #pragma unroll
    for (int r = 0; r < 8; ++r) {
      size_t idx = (size_t)(row0 + r + 8 * hi) * DIM + 16 * c + l16;
      float v = y0[idx];
      y[c][r] = v;
      out[idx] = v;
    }

  // ---- helpers -----------------------------------------------------------
  auto store_Y = [&](const v8f yy[4]) {
    __builtin_amdgcn_wave_barrier();
    #pragma unroll
    for (int c = 0; c < 4; ++c)
      #pragma unroll
      for (int r = 0; r < 8; ++r)
        Y[(r + 8 * hi) * Y_S + 16 * c + l16] = yy[c][r];
    __builtin_amdgcn_wave_barrier();
  };
  auto store_Y_axpy = [&](const v8f yy[4], const v8f kk[4], float a) {
    __builtin_amdgcn_wave_barrier();
    #pragma unroll
    for (int c = 0; c < 4; ++c)
      #pragma unroll
      for (int r = 0; r < 8; ++r)
        Y[(r + 8 * hi) * Y_S + 16 * c + l16] = yy[c][r] + a * kk[c][r];
    __builtin_amdgcn_wave_barrier();
  };

  // f(y) = tanh(y@W1 + b1) @ W2 + b2, y staged in LDS Y, result in C-layout.
  auto f_eval = [&](v8f kout[4]) {
    __builtin_amdgcn_wave_barrier();
    // Preload all 16 fp32 A-fragments of y (16x4 each): one ds_load_b64 apiece.
    v2f afr[16];
    #pragma unroll
    for (int g = 0; g < 16; ++g)
      afr[g] = *(const v2f*)&Y[l16 * Y_S + 4 * g + 2 * hi];

    v8f acc[4];
    #pragma unroll
    for (int c = 0; c < 4; ++c) {
      float b = lds.b2[16 * c + l16];
      #pragma unroll
      for (int r = 0; r < 8; ++r) acc[c][r] = b;
    }

    for (int j = 0; j < 16; j += 2) {        // hidden tiles, two chains for ILP
      v8f h0, h1;
      float bb0 = lds.b1[16 * j + l16];
      float bb1 = lds.b1[16 * j + 16 + l16];
      #pragma unroll
      for (int r = 0; r < 8; ++r) { h0[r] = bb0; h1[r] = bb1; }

      const float* w1p0 = &lds.w1t[(16 * j      + l16) * W1T_S + 2 * hi];
      const float* w1p1 = &lds.w1t[(16 * (j + 1) + l16) * W1T_S + 2 * hi];
      #pragma unroll
      for (int g = 0; g < 16; ++g) {         // K = 64 in steps of 4
        v2f bA = *(const v2f*)(w1p0 + 4 * g);
        v2f bB = *(const v2f*)(w1p1 + 4 * g);
        h0 = __builtin_amdgcn_wmma_f32_16x16x4_f32(false, afr[g], false, bA,
                                                   (short)0, h0, false, false);
        h1 = __builtin_amdgcn_wmma_f32_16x16x4_f32(false, afr[g], false, bB,
                                                   (short)0, h1, false, false);
      }
      #pragma unroll
      for (int r = 0; r < 8; ++r) { h0[r] = fast_tanh(h0[r]); h1[r] = fast_tanh(h1[r]); }

      #pragma unroll
      for (int t = 0; t < 2; ++t) {
        v8f hh = t ? h1 : h0;
        const int jj = j + t;
        // C-layout -> row-major bounce through LDS slot (same-wave DS is in-order)
        __builtin_amdgcn_wave_barrier();
        #pragma unroll
        for (int r = 0; r < 8; ++r)
          HS[(r + 8 * hi) * HS_S + l16] = hh[r];
        __builtin_amdgcn_wave_barrier();
        #pragma unroll
        for (int g2 = 0; g2 < 4; ++g2) {     // K = 16 in steps of 4
          v2f a2 = *(const v2f*)&HS[l16 * HS_S + 4 * g2 + 2 * hi];
          #pragma unroll
          for (int c = 0; c < 4; ++c) {      // 4 independent accumulators
            v2f bw = *(const v2f*)&lds.w2t[(16 * c + l16) * W2T_S
                                           + 16 * jj + 4 * g2 + 2 * hi];
            acc[c] = __builtin_amdgcn_wmma_f32_16x16x4_f32(false, a2, false, bw,
                                                           (short)0, acc[c], false, false);
          }
        }
      }
    }
    #pragma unroll
    for (int c = 0; c < 4; ++c) kout[c] = acc[c];
  };
  // ------------------------------------------------------------------------

  v8f k[4], ksum[4];
  for (int s = 0; s < TSTEPS - 1; ++s) {
    float dt = ts[s + 1] - ts[s];

    store_Y(y);
    f_eval(k);                                    // k1
    #pragma unroll
    for (int c = 0; c < 4; ++c) ksum[c] = k[c];

    store_Y_axpy(y, k, 0.5f * dt);
    f_eval(k);                                    // k2
    #pragma unroll
    for (int c = 0; c < 4; ++c) ksum[c] += 2.0f * k[c];

    store_Y_axpy(y, k, 0.5f * dt);
    f_eval(k);                                    // k3
    #pragma unroll
    for (int c = 0; c < 4; ++c) ksum[c] += 2.0f * k[c];

    store_Y_axpy(y, k, dt);
    f_eval(k);                                    // k4

    const float w = dt * (1.0f / 6.0f);
    #pragma unroll
    for (int c = 0; c < 4; ++c) y[c] = y[c] + w * (ksum[c] + k[c]);

    float* op = out + (size_t)(s + 1) * NPTS * DIM;
    #pragma unroll
    for (int c = 0; c < 4; ++c)
      #pragma unroll
      for (int r = 0; r < 8; ++r)
        op[(size_t)(row0 + r + 8 * hi) * DIM + 16 * c + l16] = y[c][r];
  }
}

extern "C" void kernel_launch(void* const* d_in, const int* in_sizes, int n_in,
                              void* d_out, int out_size, void* d_ws, size_t ws_size,
                              hipStream_t stream) {
  const float* y0 = (const float*)d_in[0];   // first_point [N,64]
  const float* ts = (const float*)d_in[1];   // time_steps  [20]
  const float* W1 = (const float*)d_in[2];   // [64,256]
  const float* b1 = (const float*)d_in[3];   // [256]
  const float* W2 = (const float*)d_in[4];   // [256,64]
  const float* b2 = (const float*)d_in[5];   // [64]
  float* out = (float*)d_out;                // [20,N,64]

  const int blocks = (TILES + WPB - 1) / WPB;   // 1563
  hipLaunchKernelGGL(node_rk4, dim3(blocks), dim3(THREADS), 0, stream,
                     y0, ts, W1, b1, W2, b2, out);
}